// NativeSparseAttention_80771154968645
// MI455X (gfx1250) — compile-verified
//
#include <hip/hip_runtime.h>
#include <math.h>

typedef __attribute__((ext_vector_type(2))) float v2f;
typedef __attribute__((ext_vector_type(8))) float v8f;

#define DEV __device__ __forceinline__

// ---------------------------------------------------------------- constants
namespace {
constexpr int S_   = 512;
constexpr int H_   = 16;
constexpr int D_   = 64;
constexpr int NB_  = 31;            // (512-32)/16 + 1
constexpr int ROWS = 2 * NB_ * H_;  // 992 (k-rows then v-rows)
constexpr int RK   = 2048;          // BLK*D
constexpr int NKCP = 256;           // nb*CBS=248 padded to 256
}

// ---------------------------------------------------------------- WMMA utils
// D = A(16x4 f32) x B(4x16 f32) + C(16x16 f32), full fp32 precision.
DEV v8f wmma_f32(v2f a, v2f b, v8f c) {
  return __builtin_amdgcn_wmma_f32_16x16x4_f32(false, a, false, b, (short)0, c,
                                               false, false);
}
// A 16x4 f32: lanes0-15 -> M=lane, VGPR0=K0,VGPR1=K1; lanes16-31 -> K2,K3
DEV v2f load_a32(const float* A, int lda, int row, int k0, int hi) {
  const float* p = A + (size_t)row * lda + k0 + (hi << 1);
  v2f a;
  a[0] = p[0];
  a[1] = p[1];
  return a;
}
// B 4x16 f32: lanes0-15 -> N=lane, VGPR0=K0,VGPR1=K1; lanes16-31 -> K2,K3
DEV v2f load_b32(const float* B, int ldb, int col, int k0, int hi) {
  const float* p = B + (size_t)(k0 + (hi << 1)) * ldb + col;
  v2f b;
  b[0] = p[0];
  b[1] = p[ldb];
  return b;
}
DEV float gelu_exact(float x) {
  return 0.5f * x * (1.0f + erff(x * 0.70710678118654752f));
}

// ---------------------------------------------------------------- prep
// rows[r][c]: r<496 -> kb row (n=r/16,h=r%16) with pos_enc; r>=496 -> vb row
__global__ void nsa_rows_kernel(const float* __restrict__ k,
                                const float* __restrict__ v,
                                const float* __restrict__ pos,
                                float* __restrict__ rows) {
  int i = blockIdx.x * 256 + threadIdx.x;
  if (i >= ROWS * RK) return;
  int r = i >> 11, c = i & 2047;
  int rr = (r < NB_ * H_) ? r : r - NB_ * H_;
  int n = rr >> 4, h = rr & 15;
  int blk = c >> 6, d = c & 63;
  int src = ((n * 16 + blk) << 10) + (h << 6) + d;   // *(H*D)=1024
  rows[i] = (r < NB_ * H_) ? (k[src] + pos[c]) : v[src];
}

// ---------------------------------------------------------------- GEMM stages
// h1 = gelu(rows @ cw1 + cb1)   M=992 K=2048 N=512
__global__ void nsa_gemm1_kernel(const float* __restrict__ A,
                                 const float* __restrict__ B,
                                 const float* __restrict__ bias,
                                 float* __restrict__ out) {
  int m0 = blockIdx.x << 4, n0 = blockIdx.y << 4;
  int lane = threadIdx.x, nl = lane & 15, hi = lane >> 4;
  v8f acc = {};
#pragma unroll 8
  for (int k0 = 0; k0 < 2048; k0 += 4) {
    v2f a = load_a32(A, 2048, m0 + nl, k0, hi);
    v2f b = load_b32(B, 512, n0 + nl, k0, hi);
    acc = wmma_f32(a, b, acc);
  }
  int col = n0 + nl;
  float bv = bias[col];
#pragma unroll
  for (int w = 0; w < 8; ++w) {
    int m = m0 + w + (hi << 3);
    out[m * 512 + col] = gelu_exact(acc[w] + bv);
  }
}

// kc/vc = h1 @ cw2 + cb2        M=992 K=512 N=512
__global__ void nsa_gemm2_kernel(const float* __restrict__ A,
                                 const float* __restrict__ B,
                                 const float* __restrict__ bias,
                                 float* __restrict__ out) {
  int m0 = blockIdx.x << 4, n0 = blockIdx.y << 4;
  int lane = threadIdx.x, nl = lane & 15, hi = lane >> 4;
  v8f acc = {};
#pragma unroll 8
  for (int k0 = 0; k0 < 512; k0 += 4) {
    v2f a = load_a32(A, 512, m0 + nl, k0, hi);
    v2f b = load_b32(B, 512, n0 + nl, k0, hi);
    acc = wmma_f32(a, b, acc);
  }
  int col = n0 + nl;
  float bv = bias[col];
#pragma unroll
  for (int w = 0; w < 8; ++w) {
    int m = m0 + w + (hi << 3);
    out[m * 512 + col] = acc[w] + bv;
  }
}

// sc[s][h][j] = scale * q[s,h,:]·kc[j/8, h, j%8, :]   (j padded to 256)
__global__ void nsa_sc_kernel(const float* __restrict__ q,
                              const float* __restrict__ kcvc,
                              float* __restrict__ sc) {
  int s0 = blockIdx.x << 4, j0 = blockIdx.y << 4, h = blockIdx.z;
  int lane = threadIdx.x, nl = lane & 15, hi = lane >> 4;
  int srow = s0 + nl;
  int j = j0 + nl;
  bool jv = j < 248;
  const float* bcol = kcvc + (size_t)(((j >> 3) << 4) + h) * 512 + ((j & 7) << 6);
  v8f acc = {};
#pragma unroll
  for (int k0 = 0; k0 < 64; k0 += 4) {
    v2f a = load_a32(q + (h << 6), 1024, srow, k0, hi);
    int kb = k0 + (hi << 1);
    v2f b;
    b[0] = jv ? bcol[kb] : 0.f;
    b[1] = jv ? bcol[kb + 1] : 0.f;
    acc = wmma_f32(a, b, acc);
  }
#pragma unroll
  for (int w = 0; w < 8; ++w) {
    int s = s0 + w + (hi << 3);
    sc[(((s << 4) + h) << 8) + j] = acc[w] * 0.125f;
  }
}

// ---------------------------------------------------------------- top-k mask
__global__ void nsa_topk_kernel(const float* __restrict__ sc,
                                unsigned* __restrict__ selmask) {
  int i = blockIdx.x * 256 + threadIdx.x;   // (s*16+h)
  if (i >= S_ * H_) return;
  const float* row = sc + (size_t)i * NKCP;
  float bs[NB_];
#pragma unroll
  for (int n = 0; n < NB_; ++n) {
    float s = 0.f;
#pragma unroll
    for (int c = 0; c < 8; ++c) s += row[n * 8 + c];
    bs[n] = s;
  }
  unsigned mask = 0;
#pragma unroll
  for (int t = 0; t < 16; ++t) {
    int best = 0;
    float bv = -INFINITY;
#pragma unroll
    for (int n = 0; n < NB_; ++n) {
      bool take = (((mask >> n) & 1u) == 0u) && (bs[n] > bv);
      bv = take ? bs[n] : bv;
      best = take ? n : best;
    }
    mask |= 1u << best;
  }
  selmask[i] = mask;
}

// ---------------------------------------------------------------- softmax weights
// pcmp = softmax over all 248; psel = softmax over selected blocks (0 elsewhere)
__global__ void nsa_weights_kernel(const float* __restrict__ sc,
                                   const unsigned* __restrict__ selmask,
                                   float* __restrict__ pcmp,
                                   float* __restrict__ psel) {
  int wid = blockIdx.x;        // (s*16+h), one wave per block
  int lane = threadIdx.x;
  const float* row = sc + (size_t)wid * NKCP;
  unsigned mask = selmask[wid];
  float val[8];
  bool  sel[8];
  float mc = -INFINITY, ms = -INFINITY;
#pragma unroll
  for (int u = 0; u < 8; ++u) {
    int j = lane + u * 32;
    float x = (j < 248) ? row[j] : -INFINITY;
    bool sl = (j < 248) && ((mask >> (j >> 3)) & 1u);
    val[u] = x;
    sel[u] = sl;
    mc = fmaxf(mc, x);
    ms = fmaxf(ms, sl ? x : -INFINITY);
  }
#pragma unroll
  for (int off = 16; off >= 1; off >>= 1) {
    mc = fmaxf(mc, __shfl_xor(mc, off, 32));
    ms = fmaxf(ms, __shfl_xor(ms, off, 32));
  }
  float ec[8], es[8], sumc = 0.f, sums = 0.f;
#pragma unroll
  for (int u = 0; u < 8; ++u) {
    int j = lane + u * 32;
    ec[u] = (j < 248) ? expf(val[u] - mc) : 0.f;
    es[u] = sel[u] ? expf(val[u] - ms) : 0.f;
    sumc += ec[u];
    sums += es[u];
  }
#pragma unroll
  for (int off = 16; off >= 1; off >>= 1) {
    sumc += __shfl_xor(sumc, off, 32);
    sums += __shfl_xor(sums, off, 32);
  }
  float rc = 1.f / sumc, rs = 1.f / sums;
#pragma unroll
  for (int u = 0; u < 8; ++u) {
    int j = lane + u * 32;
    pcmp[(size_t)wid * NKCP + j] = ec[u] * rc;
    psel[(size_t)wid * NKCP + j] = es[u] * rs;
  }
}

// ---------------------------------------------------------------- P @ Vc GEMM
// out[s,h,d] = sum_j P[s,h,j] * vc[j/8, h, j%8, d]    K=256 (weights 0 on pad)
__global__ void nsa_attn_cv_kernel(const float* __restrict__ P,
                                   const float* __restrict__ kcvc,
                                   float* __restrict__ out) {
  int s0 = blockIdx.x << 4, d0 = blockIdx.y << 4, h = blockIdx.z;
  int lane = threadIdx.x, nl = lane & 15, hi = lane >> 4;
  int srow = s0 + nl, col = d0 + nl;
  const float* A  = P + (h << 8);                       // lda = 16*256
  const float* VB = kcvc + (size_t)(NB_ * H_) * 512;    // vc rows
  v8f acc = {};
#pragma unroll 8
  for (int k0 = 0; k0 < 256; k0 += 4) {
    v2f a = load_a32(A, H_ * NKCP, srow, k0, hi);
    int kb = k0 + (hi << 1);
    int kb0 = (kb < 247) ? kb : 247;                    // weights are 0 past 247
    int kb1 = (kb + 1 < 247) ? (kb + 1) : 247;
    v2f b;
    b[0] = VB[(size_t)((((kb0 >> 3) << 4) + h)) * 512 + ((kb0 & 7) << 6) + col];
    b[1] = VB[(size_t)((((kb1 >> 3) << 4) + h)) * 512 + ((kb1 & 7) << 6) + col];
    acc = wmma_f32(a, b, acc);
  }
#pragma unroll
  for (int w = 0; w < 8; ++w) {
    int s = s0 + w + (hi << 3);
    out[(((s << 4) + h) << 6) + col] = acc[w];
  }
}

// ---------------------------------------------------------------- window attn
__global__ void nsa_win_kernel(const float* __restrict__ q,
                               const float* __restrict__ k,
                               const float* __restrict__ v,
                               float* __restrict__ out) {
  int wid = blockIdx.x;          // s*16 + h
  int t = wid >> 4, h = wid & 15;
  int lane = threadIdx.x;
  int start = (t > 63) ? (t - 63) : 0;
  const float* qrow = q + (t << 10) + (h << 6);
  int p0 = start + lane, p1 = start + lane + 32;
  const float* k0p = k + (p0 << 10) + (h << 6);
  const float* k1p = k + (p1 << 10) + (h << 6);
  float s0 = 0.f, s1 = 0.f;
  for (int d = 0; d < 64; ++d) {
    float qd = qrow[d];
    s0 += qd * k0p[d];
    s1 += qd * k1p[d];
  }
  bool v0 = p0 <= t, v1 = p1 <= t;
  s0 = v0 ? s0 * 0.125f : 0.f;   // invalid slots: kk==0 -> score exactly 0
  s1 = v1 ? s1 * 0.125f : 0.f;
  float m = fmaxf(s0, s1);
#pragma unroll
  for (int off = 16; off >= 1; off >>= 1) m = fmaxf(m, __shfl_xor(m, off, 32));
  float e0 = expf(s0 - m), e1 = expf(s1 - m);
  float sum = e0 + e1;
#pragma unroll
  for (int off = 16; off >= 1; off >>= 1) sum += __shfl_xor(sum, off, 32);
  float w0 = e0 / sum, w1 = e1 / sum;
  float o0 = 0.f, o1 = 0.f;
  for (int j = 0; j < 64; ++j) {
    int pj = start + j;
    if (pj > t) continue;                         // vv==0 there (uniform branch)
    float wj = (j < 32) ? __shfl(w0, j, 32) : __shfl(w1, j - 32, 32);
    const float* vp = v + (pj << 10) + (h << 6);
    o0 += wj * vp[lane];
    o1 += wj * vp[lane + 32];
  }
  out[(wid << 6) + lane]      = o0;
  out[(wid << 6) + lane + 32] = o1;
}

// ---------------------------------------------------------------- gate
__global__ void nsa_gate_kernel(const float* __restrict__ q,
                                const float* __restrict__ gw1,
                                const float* __restrict__ gb1,
                                const float* __restrict__ gw2,
                                const float* __restrict__ gb2,
                                float* __restrict__ g) {
  int wid = blockIdx.x;          // s*16 + h
  int t = wid >> 4, h = wid & 15;
  int lane = threadIdx.x;
  const float* qrow = q + (t << 10) + (h << 6);
  int o0 = lane, o1 = lane + 32;
  float h0 = gb1[o0], h1 = gb1[o1];
  for (int d = 0; d < 64; ++d) {
    float qd = qrow[d];
    h0 += qd * gw1[d * 64 + o0];
    h1 += qd * gw1[d * 64 + o1];
  }
  h0 = gelu_exact(h0);
  h1 = gelu_exact(h1);
  float gi[3];
#pragma unroll
  for (int i = 0; i < 3; ++i) {
    float p = h0 * gw2[o0 * 3 + i] + h1 * gw2[o1 * 3 + i];
#pragma unroll
    for (int off = 16; off >= 1; off >>= 1) p += __shfl_xor(p, off, 32);
    gi[i] = 1.f / (1.f + expf(-(p + gb2[i])));
  }
  float s = gi[0] + gi[1] + gi[2] + 1e-6f;
  if (lane == 0) {
    g[wid * 4 + 0] = gi[0] / s;
    g[wid * 4 + 1] = gi[1] / s;
    g[wid * 4 + 2] = gi[2] / s;
  }
}

// ---------------------------------------------------------------- combine
__global__ void nsa_combine_kernel(const float* __restrict__ oc,
                                   const float* __restrict__ os,
                                   const float* __restrict__ ow,
                                   const float* __restrict__ g,
                                   float* __restrict__ opre) {
  int i = blockIdx.x * 256 + threadIdx.x;    // (s*16+h)*64 + d == s*1024+h*64+d
  if (i >= S_ * H_ * D_) return;
  int sh = i >> 6;
  opre[i] = g[sh * 4 + 0] * oc[i] + g[sh * 4 + 1] * os[i] + g[sh * 4 + 2] * ow[i];
}

// final: out = opre @ wo + bo     M=512 K=1024 N=1024
__global__ void nsa_gemm_out_kernel(const float* __restrict__ A,
                                    const float* __restrict__ B,
                                    const float* __restrict__ bias,
                                    float* __restrict__ out) {
  int m0 = blockIdx.x << 4, n0 = blockIdx.y << 4;
  int lane = threadIdx.x, nl = lane & 15, hi = lane >> 4;
  v8f acc = {};
#pragma unroll 8
  for (int k0 = 0; k0 < 1024; k0 += 4) {
    v2f a = load_a32(A, 1024, m0 + nl, k0, hi);
    v2f b = load_b32(B, 1024, n0 + nl, k0, hi);
    acc = wmma_f32(a, b, acc);
  }
  int col = n0 + nl;
  float bv = bias[col];
#pragma unroll
  for (int w = 0; w < 8; ++w) {
    int m = m0 + w + (hi << 3);
    out[m * 1024 + col] = acc[w] + bv;
  }
}

// ---------------------------------------------------------------- launcher
extern "C" void kernel_launch(void* const* d_in, const int* in_sizes, int n_in,
                              void* d_out, int out_size, void* d_ws, size_t ws_size,
                              hipStream_t stream) {
  (void)in_sizes; (void)n_in; (void)out_size; (void)ws_size;
  const float* q   = (const float*)d_in[0];
  const float* k   = (const float*)d_in[1];
  const float* v   = (const float*)d_in[2];
  const float* pos = (const float*)d_in[3];
  const float* cw1 = (const float*)d_in[4];
  const float* cb1 = (const float*)d_in[5];
  const float* cw2 = (const float*)d_in[6];
  const float* cb2 = (const float*)d_in[7];
  const float* gw1 = (const float*)d_in[8];
  const float* gb1 = (const float*)d_in[9];
  const float* gw2 = (const float*)d_in[10];
  const float* gb2 = (const float*)d_in[11];
  const float* wo  = (const float*)d_in[12];
  const float* bo  = (const float*)d_in[13];
  float* out = (float*)d_out;

  char* base = (char*)d_ws;
  size_t off = 0;
  auto alloc = [&](size_t bytes) {
    char* p = base + off;
    off += (bytes + 255) & ~(size_t)255;
    return p;
  };
  float*    rows   = (float*)alloc((size_t)ROWS * RK * 4);     // 992x2048
  float*    h1     = (float*)alloc((size_t)ROWS * 512 * 4);
  float*    kcvc   = (float*)alloc((size_t)ROWS * 512 * 4);    // kc rows 0..495, vc 496..991
  float*    scbuf  = (float*)alloc((size_t)S_ * H_ * NKCP * 4);
  unsigned* selmsk = (unsigned*)alloc((size_t)S_ * H_ * 4);
  float*    pcmp   = (float*)alloc((size_t)S_ * H_ * NKCP * 4);
  float*    psel   = (float*)alloc((size_t)S_ * H_ * NKCP * 4);
  float*    ocmp   = (float*)alloc((size_t)S_ * H_ * D_ * 4);
  float*    osel   = (float*)alloc((size_t)S_ * H_ * D_ * 4);
  float*    owin   = (float*)alloc((size_t)S_ * H_ * D_ * 4);
  float*    gbuf   = (float*)alloc((size_t)S_ * H_ * 4 * 4);
  float*    opre   = (float*)alloc((size_t)512 * 1024 * 4);

  // prep: blockified rows (k + pos_enc, v)
  nsa_rows_kernel<<<(ROWS * RK + 255) / 256, 256, 0, stream>>>(k, v, pos, rows);

  // compression MLP (weights consumed directly in fp32)
  nsa_gemm1_kernel<<<dim3(ROWS / 16, 512 / 16), 32, 0, stream>>>(rows, cw1, cb1, h1);
  nsa_gemm2_kernel<<<dim3(ROWS / 16, 512 / 16), 32, 0, stream>>>(h1, cw2, cb2, kcvc);

  // scores, top-k, softmax weights
  nsa_sc_kernel<<<dim3(S_ / 16, NKCP / 16, H_), 32, 0, stream>>>(q, kcvc, scbuf);
  nsa_topk_kernel<<<(S_ * H_ + 255) / 256, 256, 0, stream>>>(scbuf, selmsk);
  nsa_weights_kernel<<<S_ * H_, 32, 0, stream>>>(scbuf, selmsk, pcmp, psel);

  // compressed + selected attention outputs (P @ Vc)
  nsa_attn_cv_kernel<<<dim3(S_ / 16, D_ / 16, H_), 32, 0, stream>>>(pcmp, kcvc, ocmp);
  nsa_attn_cv_kernel<<<dim3(S_ / 16, D_ / 16, H_), 32, 0, stream>>>(psel, kcvc, osel);

  // sliding-window attention + gate
  nsa_win_kernel<<<S_ * H_, 32, 0, stream>>>(q, k, v, owin);
  nsa_gate_kernel<<<S_ * H_, 32, 0, stream>>>(q, gw1, gb1, gw2, gb2, gbuf);

  // combine + output projection
  nsa_combine_kernel<<<(S_ * H_ * D_ + 255) / 256, 256, 0, stream>>>(ocmp, osel, owin, gbuf, opre);
  nsa_gemm_out_kernel<<<dim3(512 / 16, 1024 / 16), 32, 0, stream>>>(opre, wo, bo, out);
}